// EM_online_42889543418346
// MI455X (gfx1250) — compile-verified
//
#include <hip/hip_runtime.h>
#include <hip/hip_bf16.h>
#include <stdint.h>

#define NROW 200000
#define RDIM 32
#define BATCH 131072
#define STEP 0.01f

// ---------------------------------------------------------------------------
// Kernel 1: one wave32 per batch sample; lane == r (R == 32 exactly).
// L is staged into LDS with the CDNA5 async global->LDS path (ASYNCcnt).
// Segment sums are accumulated directly into d_out via global_atomic_add_f32.
// ---------------------------------------------------------------------------
__global__ void em_scatter(const float* __restrict__ U0,
                           const float* __restrict__ U1,
                           const float* __restrict__ U2,
                           const float* __restrict__ L,
                           const float* __restrict__ vals,
                           const int*   __restrict__ ent,
                           float* __restrict__ accS,   // d_out[0 .. 3*NROW*R)
                           float* __restrict__ accT,   // d_out[3*NROW*R ..)
                           unsigned int* __restrict__ touched) // ws, 3*NROW
{
    __shared__ float sL[RDIM];
    const int lane = threadIdx.x & 31;

    // Stage L (32 floats) into LDS via async global->LDS DMA (gfx1250 path).
    if (threadIdx.x < 32) {
        uint32_t lds_addr = (uint32_t)(uintptr_t)(&sL[lane]);
        uint64_t gaddr    = (uint64_t)(uintptr_t)(L + lane);
        asm volatile("global_load_async_to_lds_b32 %0, %1, off"
                     :: "v"(lds_addr), "v"(gaddr) : "memory");
        asm volatile("s_wait_asynccnt 0" ::: "memory");
    }
    __syncthreads();

    const int b = (int)((blockIdx.x * blockDim.x + threadIdx.x) >> 5);
    if (b >= BATCH) return;

    const int i0 = ent[b * 3 + 0];
    const int i1 = ent[b * 3 + 1];
    const int i2 = ent[b * 3 + 2];

    // Coalesced 128B row gathers: lane r reads element r of each row.
    const float g0 = U0[(size_t)i0 * RDIM + lane];
    const float g1 = U1[(size_t)i1 * RDIM + lane];
    const float g2 = U2[(size_t)i2 * RDIM + lane];
    const float Lr = sL[lane];

    const float p  = g0 * g1 * g2;
    const float lp = Lr * p;

    // phi = sum_r L[r]*p[r] : wave32 butterfly reduction.
    float phi = lp;
    #pragma unroll
    for (int off = 16; off > 0; off >>= 1)
        phi += __shfl_xor(phi, off, 32);

    const float wi    = 0.5f / phi * tanhf(0.5f * phi);
    const float ki    = vals[b] - 0.5f;
    const float dvec  = phi - lp;          // same for all 3 dims
    const float tcoef = ki - dvec * wi;

    const size_t plane = (size_t)NROW * RDIM;

    // d = 0 : prod_others = g1*g2, scatter to row i0
    {
        const float c = Lr * (g1 * g2);
        atomicAdd(&accS[0 * plane + (size_t)i0 * RDIM + lane], c * c * wi);
        atomicAdd(&accT[0 * plane + (size_t)i0 * RDIM + lane], tcoef * c);
        if (lane == 0) touched[0 * NROW + i0] = 1u;  // benign same-value race
    }
    // d = 1 : prod_others = g0*g2, scatter to row i1
    {
        const float c = Lr * (g0 * g2);
        atomicAdd(&accS[1 * plane + (size_t)i1 * RDIM + lane], c * c * wi);
        atomicAdd(&accT[1 * plane + (size_t)i1 * RDIM + lane], tcoef * c);
        if (lane == 0) touched[1 * NROW + i1] = 1u;
    }
    // d = 2 : prod_others = g0*g1, scatter to row i2
    {
        const float c = Lr * (g0 * g1);
        atomicAdd(&accS[2 * plane + (size_t)i2 * RDIM + lane], c * c * wi);
        atomicAdd(&accT[2 * plane + (size_t)i2 * RDIM + lane], tcoef * c);
        if (lane == 0) touched[2 * NROW + i2] = 1u;
    }
}

// ---------------------------------------------------------------------------
// Kernel 2: streaming float4 finalize, in place on d_out.
//   out_S = touched ? 0.99*S + 0.01*accS : S   (same for T)
// ---------------------------------------------------------------------------
__global__ void em_finalize(const float* __restrict__ S0,
                            const float* __restrict__ S1,
                            const float* __restrict__ S2,
                            const float* __restrict__ T0,
                            const float* __restrict__ T1,
                            const float* __restrict__ T2,
                            const unsigned int* __restrict__ touched,
                            float* __restrict__ out)
{
    const size_t plane = (size_t)NROW * RDIM;       // 6.4M elements
    const size_t nvec  = 3 * plane / 4;             // float4 chunks of S-part
    const size_t tid   = (size_t)blockIdx.x * blockDim.x + threadIdx.x;
    if (tid >= nvec) return;

    const size_t e   = tid * 4;                     // flat (3,NROW,R) element
    const int    d   = (int)(e / plane);
    const size_t rem = e - (size_t)d * plane;       // index within (NROW,R)
    const int    row = (int)(rem / RDIM);

    const bool tch = touched[(size_t)d * NROW + row] != 0u;

    const float* Sd = (d == 0) ? S0 : ((d == 1) ? S1 : S2);
    const float* Td = (d == 0) ? T0 : ((d == 1) ? T1 : T2);

    const float4 s  = *(const float4*)(Sd + rem);
    const float4 t  = *(const float4*)(Td + rem);
    float4 as = *(const float4*)(out + e);                   // accS
    float4 at = *(const float4*)(out + 3 * plane + e);       // accT

    const float a = 1.0f - STEP, bq = STEP;
    float4 os, ot;
    if (tch) {
        os.x = a * s.x + bq * as.x;  os.y = a * s.y + bq * as.y;
        os.z = a * s.z + bq * as.z;  os.w = a * s.w + bq * as.w;
        ot.x = a * t.x + bq * at.x;  ot.y = a * t.y + bq * at.y;
        ot.z = a * t.z + bq * at.z;  ot.w = a * t.w + bq * at.w;
    } else {
        os = s; ot = t;
    }
    *(float4*)(out + e)             = os;
    *(float4*)(out + 3 * plane + e) = ot;
}

// ---------------------------------------------------------------------------
// Launch. Input order = setup_inputs() dict order:
//   0:U0 1:S0 2:T0 3:U1 4:S1 5:T1 6:U2 7:S2 8:T2 9:L 10:batch_vals 11:batch_entries
// Output: stack([S_new, T_new]) -> (2,3,NROW,R) flat f32.
// ---------------------------------------------------------------------------
extern "C" void kernel_launch(void* const* d_in, const int* in_sizes, int n_in,
                              void* d_out, int out_size, void* d_ws, size_t ws_size,
                              hipStream_t stream) {
    const float* U0 = (const float*)d_in[0];
    const float* S0 = (const float*)d_in[1];
    const float* T0 = (const float*)d_in[2];
    const float* U1 = (const float*)d_in[3];
    const float* S1 = (const float*)d_in[4];
    const float* T1 = (const float*)d_in[5];
    const float* U2 = (const float*)d_in[6];
    const float* S2 = (const float*)d_in[7];
    const float* T2 = (const float*)d_in[8];
    const float* L  = (const float*)d_in[9];
    const float* bv = (const float*)d_in[10];
    const int*   be = (const int*)d_in[11];

    float* out = (float*)d_out;
    unsigned int* touched = (unsigned int*)d_ws;

    const size_t plane = (size_t)NROW * RDIM;

    // Zero accumulators (d_out doubles as the segment-sum buffer) + flags.
    hipMemsetAsync(d_out, 0, 2 * 3 * plane * sizeof(float), stream);
    hipMemsetAsync(d_ws, 0, 3 * (size_t)NROW * sizeof(unsigned int), stream);

    // 1 wave per sample, 8 waves (256 threads) per block.
    const int blocks_scatter = BATCH / 8;          // 16384
    em_scatter<<<blocks_scatter, 256, 0, stream>>>(
        U0, U1, U2, L, bv, be, out, out + 3 * plane, touched);

    // float4-vectorized finalize over 3*NROW*R elements.
    const size_t nvec = 3 * plane / 4;             // 4,800,000
    const int blocks_fin = (int)((nvec + 255) / 256);
    em_finalize<<<blocks_fin, 256, 0, stream>>>(
        S0, S1, S2, T0, T1, T2, touched, out);
}